// AdditiveAttention_79096117723792
// MI455X (gfx1250) — compile-verified
//
#include <hip/hip_runtime.h>

// ---------------------------------------------------------------------------
// AdditiveAttention (Bahdanau-style) for MI455X / gfx1250, wave32.
//   out = softmax_k( Wv . tanh(q Wq^T [+broadcast] k Wk^T) ) @ v
// Outputs concatenated: out [2,8,512,64] f32, then attn [2,8,512,512] f32.
//
// Projections and attn@v run on V_WMMA_F32_16X16X4_F32. The tanh score loop
// (268M tanh -> TRANS-pipe bound) is register-blocked 4 q-rows x 32 k-cols
// per wave so each kp/Wv value is loaded once per 4 tanh, via ds_load_b64.
// ---------------------------------------------------------------------------

typedef float v2f __attribute__((ext_vector_type(2)));
typedef float v8f __attribute__((ext_vector_type(8)));

#define B_  2
#define H_  8
#define T_  512
#define D_  64
#define QB  32            // q rows per workgroup
#define KT  256           // k rows per LDS tile (2 tiles cover T)
#define KP_STRIDE 66      // 66 % 64 == 2 -> conflict-free lane reads & frag stores
#define QP_STRIDE 66
#define AT_STRIDE 516     // 516 % 64 == 4 -> conflict-free WMMA A-frag reads

__device__ __forceinline__ float fast_tanh(float x) {
#if __has_builtin(__builtin_amdgcn_tanhf)
  return __builtin_amdgcn_tanhf(x);     // v_tanh_f32 (quarter-rate TRANS)
#else
  return tanhf(x);
#endif
}

__device__ __forceinline__ float wave_max(float v) {
  #pragma unroll
  for (int off = 16; off > 0; off >>= 1) v = fmaxf(v, __shfl_xor(v, off, 32));
  return v;
}
__device__ __forceinline__ float wave_sum(float v) {
  #pragma unroll
  for (int off = 16; off > 0; off >>= 1) v += __shfl_xor(v, off, 32);
  return v;
}

__global__ __launch_bounds__(256)
void addattn_kernel(const float* __restrict__ q,
                    const float* __restrict__ k,
                    const float* __restrict__ v,
                    const unsigned char* __restrict__ mask,  // bool, 1 byte
                    const float* __restrict__ Wq,            // [64,64]
                    const float* __restrict__ Wk,            // [64,64]
                    const float* __restrict__ Wv,            // [64]
                    float* __restrict__ out,                 // [16,512,64]
                    float* __restrict__ attn_out)            // [16,512,512]
{
  __shared__ float kp_s[KT][KP_STRIDE];   // k_proj tile, row-major [k][a]
  __shared__ float qp_s[QB][QP_STRIDE];   // q_proj block, [q][a]
  __shared__ float at_s[QB][AT_STRIDE];   // scores -> attn, [q][k]
  __shared__ float wv_s[D_];

  const int head  = blockIdx.y;           // 0..15  (b*H+h)
  const int qrow0 = blockIdx.x * QB;      // 0,32,...,480

  const int tid  = threadIdx.x;
  const int wave = tid >> 5;              // 0..7
  const int lane = tid & 31;
  const int half = lane >> 4;             // 0/1
  const int m16  = lane & 15;             // 0..15

  const float* qh = q + (size_t)head * T_ * D_;
  const float* kh = k + (size_t)head * T_ * D_;
  const float* vh = v + (size_t)head * T_ * D_;
  const unsigned char* mh = mask + (size_t)head * T_ * T_;
  float* oh = out      + (size_t)head * T_ * D_;
  float* ah = attn_out + (size_t)head * T_ * T_;

  if (tid < D_) wv_s[tid] = Wv[tid];

  // Prefetch the v head tile (128KB) into cache while the long tanh phase
  // runs; phase 3 reads it as WMMA B-fragments. global_prefetch_b8, no cnt.
  {
    const char* vp = (const char*)vh + (size_t)tid * 512;
    #pragma unroll
    for (int j = 0; j < 4; ++j) __builtin_prefetch(vp + j * 128, 0, 1);
  }

  // ---- Phase 0: q_proj (QB x 64) = q_blk (QB x 64) @ Wq^T via WMMA f32 16x16x4.
  // A[m][kk]: lane m16 holds row m; vgpr0/1 = K (4s+2*half)+{0,1}.
  // B[kk][n] = Wq[n][kk]: lane m16 holds column n; vgpr0/1 = same K pair.
  {
    const int mi = wave >> 2;             // 0..1  (q sub-tile)
    const int ni = wave & 3;              // 0..3  (a sub-tile)
    v8f c = {};
    #pragma unroll
    for (int s = 0; s < 16; ++s) {
      const int kk = s * 4 + half * 2;
      v2f a, b;
      const float* ap = qh + (size_t)(qrow0 + mi * 16 + m16) * D_ + kk;
      a.x = ap[0]; a.y = ap[1];
      const float* bp = Wq + (size_t)(ni * 16 + m16) * D_ + kk;
      b.x = bp[0]; b.y = bp[1];
      c = __builtin_amdgcn_wmma_f32_16x16x4_f32(false, a, false, b,
                                                (short)0, c, false, false);
    }
    #pragma unroll
    for (int r = 0; r < 8; ++r)           // D: vgpr r -> row r+8*half, col m16
      qp_s[mi * 16 + r + half * 8][ni * 16 + m16] = c[r];
  }
  __syncthreads();

  const int wq0 = wave * 4;               // 4 q-rows owned per wave

  for (int kt = 0; kt < T_ / KT; ++kt) {
    // ---- Phase 1a: k_proj tile (KT x 64) via WMMA; 64 tiles, 8 per wave.
    #pragma unroll
    for (int t = 0; t < 8; ++t) {
      const int tile = wave * 8 + t;      // 0..63
      const int mi = tile >> 2;           // 0..15 (16 rows each)
      const int ni = tile & 3;            // 0..3
      v8f c = {};
      #pragma unroll
      for (int s = 0; s < 16; ++s) {
        const int kkd = s * 4 + half * 2;
        v2f a, b;
        const float* ap = kh + (size_t)(kt * KT + mi * 16 + m16) * D_ + kkd;
        a.x = ap[0]; a.y = ap[1];
        const float* bp = Wk + (size_t)(ni * 16 + m16) * D_ + kkd;
        b.x = bp[0]; b.y = bp[1];
        c = __builtin_amdgcn_wmma_f32_16x16x4_f32(false, a, false, b,
                                                  (short)0, c, false, false);
      }
      #pragma unroll
      for (int r = 0; r < 8; ++r)
        kp_s[mi * 16 + r + half * 8][ni * 16 + m16] = c[r];
    }
    __syncthreads();

    // ---- Phase 1b: additive tanh scores (the hot loop).
    // lane -> k column; 4 q-rows register-blocked so each kp/Wv float2 is
    // loaded once per 8 tanh. All LDS reads are b64 and conflict-free.
    #pragma unroll
    for (int kc = 0; kc < KT / 32; ++kc) {
      const int kk = kc * 32 + lane;
      float acc0 = 0.f, acc1 = 0.f, acc2 = 0.f, acc3 = 0.f;
      #pragma unroll 4
      for (int a = 0; a < D_; a += 2) {
        const v2f kp = *(const v2f*)&kp_s[kk][a];
        const v2f wv = *(const v2f*)&wv_s[a];
        const v2f q0 = *(const v2f*)&qp_s[wq0 + 0][a];
        const v2f q1 = *(const v2f*)&qp_s[wq0 + 1][a];
        const v2f q2 = *(const v2f*)&qp_s[wq0 + 2][a];
        const v2f q3 = *(const v2f*)&qp_s[wq0 + 3][a];
        acc0 = fmaf(wv.x, fast_tanh(q0.x + kp.x),
               fmaf(wv.y, fast_tanh(q0.y + kp.y), acc0));
        acc1 = fmaf(wv.x, fast_tanh(q1.x + kp.x),
               fmaf(wv.y, fast_tanh(q1.y + kp.y), acc1));
        acc2 = fmaf(wv.x, fast_tanh(q2.x + kp.x),
               fmaf(wv.y, fast_tanh(q2.y + kp.y), acc2));
        acc3 = fmaf(wv.x, fast_tanh(q3.x + kp.x),
               fmaf(wv.y, fast_tanh(q3.y + kp.y), acc3));
      }
      const int kabs = kt * KT + kk;
      const size_t mrow = (size_t)(qrow0 + wq0) * T_ + kabs;
      at_s[wq0 + 0][kabs] = mh[mrow         ] ? acc0 : -1e9f;
      at_s[wq0 + 1][kabs] = mh[mrow +     T_] ? acc1 : -1e9f;
      at_s[wq0 + 2][kabs] = mh[mrow + 2 * T_] ? acc2 : -1e9f;
      at_s[wq0 + 3][kabs] = mh[mrow + 3 * T_] ? acc3 : -1e9f;
    }
    __syncthreads();                      // protect kp_s reuse next kt
  }

  // ---- Phase 2: softmax per q-row (wave-local, wave32 shfl reductions),
  // write attn to LDS (for WMMA) and to global (coalesced 128B/wave).
  for (int qr = wq0; qr < wq0 + 4; ++qr) {
    float mx = -3.402823466e38f;
    #pragma unroll
    for (int j = 0; j < T_ / 32; ++j)
      mx = fmaxf(mx, at_s[qr][j * 32 + lane]);
    mx = wave_max(mx);
    float e[T_ / 32];
    float sum = 0.f;
    #pragma unroll
    for (int j = 0; j < T_ / 32; ++j) {
      e[j] = __expf(at_s[qr][j * 32 + lane] - mx);
      sum += e[j];
    }
    sum = wave_sum(sum);
    const float inv = 1.f / sum;
    float* arow = ah + (size_t)(qrow0 + qr) * T_;
    #pragma unroll
    for (int j = 0; j < T_ / 32; ++j) {
      const float p = e[j] * inv;
      at_s[qr][j * 32 + lane] = p;
      arow[j * 32 + lane] = p;
    }
  }
  __syncthreads();

  // ---- Phase 3: out_blk (QB x 64) = attn_blk (QB x 512) @ v (512 x 64).
  // 2x4 = 8 tiles of 16x16, one per wave; 128 WMMA K-steps each.
  {
    const int mi = wave >> 2;             // 0..1
    const int ni = wave & 3;              // 0..3
    v8f c = {};
    #pragma unroll 4
    for (int s = 0; s < T_ / 4; ++s) {
      const int kk = s * 4 + half * 2;
      const v2f a = *(const v2f*)&at_s[mi * 16 + m16][kk];  // ds_load_b64
      v2f b;
      const float* bp = vh + (size_t)kk * D_ + ni * 16 + m16;
      b.x = bp[0];                        // B[kk][n]   (coalesced over lanes)
      b.y = bp[D_];                       // B[kk+1][n]
      c = __builtin_amdgcn_wmma_f32_16x16x4_f32(false, a, false, b,
                                                (short)0, c, false, false);
    }
    #pragma unroll
    for (int r = 0; r < 8; ++r)
      oh[(size_t)(qrow0 + mi * 16 + r + half * 8) * D_ + ni * 16 + m16] = c[r];
  }
}

extern "C" void kernel_launch(void* const* d_in, const int* in_sizes, int n_in,
                              void* d_out, int out_size, void* d_ws, size_t ws_size,
                              hipStream_t stream) {
  (void)in_sizes; (void)n_in; (void)d_ws; (void)ws_size; (void)out_size;
  const float*         q    = (const float*)d_in[0];
  const float*         k    = (const float*)d_in[1];
  const float*         v    = (const float*)d_in[2];
  const unsigned char* mask = (const unsigned char*)d_in[3];  // bool -> 1 byte
  const float*         Wq   = (const float*)d_in[4];
  const float*         Wk   = (const float*)d_in[5];
  const float*         Wv   = (const float*)d_in[6];

  float* out  = (float*)d_out;                            // [2,8,512,64]
  float* attn = out + (size_t)B_ * H_ * T_ * D_;          // [2,8,512,512]

  dim3 grid(T_ / QB, B_ * H_);                            // 16 x 16 blocks
  addattn_kernel<<<grid, 256, 0, stream>>>(q, k, v, mask, Wq, Wk, Wv, out, attn);
}